// LIFSpike_43542378447546
// MI455X (gfx1250) — compile-verified
//
#include <hip/hip_runtime.h>

// LIF spike scan: x[B,T,C,H,W] fp32 -> spikes[B,T,C,H,W] fp32
// B=32, T=8, C=128, H=32, W=32. Recurrence over T per (b,c,h,w) position.
// Memory-bound streaming kernel: 268 MB total traffic, ~11.5 us floor @ 23.3 TB/s.

constexpr int Bq   = 32;
constexpr int Tq   = 8;
constexpr int CHW  = 128 * 32 * 32;   // 131072 elements: T-axis stride
constexpr int TCHW = Tq * CHW;        // 1048576 elements: B-axis stride
constexpr int NPOS = Bq * CHW;        // 4194304 independent recurrences
constexpr int NV4  = NPOS / 4;        // 1048576 float4 lanes of work
constexpr int CHW4 = CHW / 4;         // 32768 float4 groups per batch image

constexpr float TAU    = 0.5f;
constexpr float THRESH = 1.0f;

typedef __attribute__((ext_vector_type(4))) float v4f;

__global__ __launch_bounds__(256) void lif_scan_kernel(
    const float* __restrict__ x, float* __restrict__ out) {
  const int tid = blockIdx.x * blockDim.x + threadIdx.x;   // [0, NV4)
  // Decompose into (batch, within-image float4 index). CHW4 = 2^15.
  const int b     = tid >> 15;
  const int inner = tid & (CHW4 - 1);
  const int base  = b * TCHW + inner * 4;                  // element offset, < 2^25

  // Issue all 8 time-slice loads up front: 8 outstanding 128b NT loads per
  // lane -> deep MLP; data streams through (NT) without thrashing 192MB L2.
  v4f xv[Tq];
#pragma unroll
  for (int t = 0; t < Tq; ++t) {
    xv[t] = __builtin_nontemporal_load(
        reinterpret_cast<const v4f*>(x + base + t * CHW));
  }

  // Serial LIF scan over T (dependent chain is cheap VALU; store each spike
  // vector immediately so we never hold more than xv[] + mem live).
  v4f mem = (v4f)0.0f;
#pragma unroll
  for (int t = 0; t < Tq; ++t) {
    v4f spike;
#pragma unroll
    for (int c = 0; c < 4; ++c) {
      float m = mem[c] * TAU + xv[t][c];
      float s = (m > THRESH) ? 1.0f : 0.0f;   // zif(m - THRESH)
      mem[c]  = (s > 0.0f) ? 0.0f : m;        // hard reset where spiked
      spike[c] = s;
    }
    __builtin_nontemporal_store(spike,
        reinterpret_cast<v4f*>(out + base + t * CHW));
  }
}

extern "C" void kernel_launch(void* const* d_in, const int* in_sizes, int n_in,
                              void* d_out, int out_size, void* d_ws, size_t ws_size,
                              hipStream_t stream) {
  const float* x = reinterpret_cast<const float*>(d_in[0]);
  float* out     = reinterpret_cast<float*>(d_out);

  constexpr int BLOCK = 256;                 // 8 wave32s per workgroup
  constexpr int GRID  = NV4 / BLOCK;         // 4096 blocks, exact cover
  lif_scan_kernel<<<GRID, BLOCK, 0, stream>>>(x, out);
}